// GCN_node_classification_53884659695768
// MI455X (gfx1250) — compile-verified
//
#include <hip/hip_runtime.h>
#include <hip/hip_bf16.h>
#include <math.h>

// ---------------------------------------------------------------------------
// GCN node classification, MI455X (gfx1250, wave32).
// fp32 WMMA (v_wmma_f32_16x16x4_f32) for all GEMMs with pre-packed,
// zero-padded B fragments (branch-free inner loop); coalesced
// global_atomic_add_f32 scatter for the GCN aggregation.
// ---------------------------------------------------------------------------

typedef float v2f __attribute__((ext_vector_type(2)));
typedef float v8f __attribute__((ext_vector_type(8)));

#define DIN 128
#define DH 256
#define DOUT 40
#define DOUTP 48  // DOUT padded to a multiple of 16
#define NLAYERS 4

// ---------------------------------------------------------------------------
// small elementwise kernels
// ---------------------------------------------------------------------------
__global__ void zero_kernel(float* __restrict__ p, int n) {
  int i = blockIdx.x * blockDim.x + threadIdx.x;
  if (i < n) p[i] = 0.0f;
}

__global__ void deg_kernel(const long long* __restrict__ dst,
                           float* __restrict__ deg, int nE) {
  int e = blockIdx.x * blockDim.x + threadIdx.x;
  if (e < nE) atomicAdd(&deg[(int)dst[e]], 1.0f);
}

__global__ void dinv_kernel(float* __restrict__ d, int n) {
  int i = blockIdx.x * blockDim.x + threadIdx.x;
  if (i < n) d[i] = rsqrtf(d[i] + 1.0f);  // +1 self loop
}

__global__ void coef_kernel(const long long* __restrict__ src,
                            const long long* __restrict__ dst,
                            const float* __restrict__ dinv,
                            float* __restrict__ coef, int nE) {
  int e = blockIdx.x * blockDim.x + threadIdx.x;
  if (e < nE) coef[e] = dinv[(int)src[e]] * dinv[(int)dst[e]];
}

// out[n,d] = h[n,d] * dinv[n]^2   (self-loop term; initializes the scatter
// accumulation buffer)
__global__ void agg_init_kernel(const float* __restrict__ h,
                                const float* __restrict__ dinv,
                                float* __restrict__ out, int total, int logd) {
  int i = blockIdx.x * blockDim.x + threadIdx.x;
  if (i >= total) return;
  float di = dinv[i >> logd];
  out[i] = h[i] * di * di;
}

// ---------------------------------------------------------------------------
// Repack W[K,Mreal] (row-major) into per-lane WMMA B-fragment order,
// zero-padded to Mpad columns:
//   flat = ((t*(K/4) + k4)*2 + kh)*32 + m*2 + j
// where the source element is W[(k4*4 + 2*kh + j)*Mreal + t*16 + m].
// A lane (n = m, half = kh) then reads its (b.x,b.y) pair for k-step k4 as
// ONE contiguous 8-byte load at immediate offset k4*256B.
// ---------------------------------------------------------------------------
__global__ void packB_kernel(const float* __restrict__ W,
                             float* __restrict__ Bt, int K, int Mreal,
                             int Mpad) {
  int i = blockIdx.x * blockDim.x + threadIdx.x;
  if (i >= K * Mpad) return;
  int j = i & 1;
  int m = (i >> 1) & 15;
  int kh = (i >> 5) & 1;
  int rest = i >> 6;  // t*(K/4) + k4
  int K4 = K >> 2;
  int k4 = rest % K4;
  int t = rest / K4;
  int kk = k4 * 4 + 2 * kh + j;
  int col = t * 16 + m;
  Bt[i] = (col < Mreal) ? W[kk * Mreal + col] : 0.0f;
}

__global__ void padbias_kernel(const float* __restrict__ b,
                               float* __restrict__ bp, int Mreal, int Mpad) {
  int i = blockIdx.x * blockDim.x + threadIdx.x;
  if (i < Mpad) bp[i] = (i < Mreal) ? b[i] : 0.0f;
}

// ---------------------------------------------------------------------------
// edge scatter: out[dst,:] += h[src,:] * coef[e]
// blockDim == D so lane d handles feature d -> fully coalesced gather row
// reads and coalesced global_atomic_add_f32 scatter.
// ---------------------------------------------------------------------------
template <int D>
__global__ __launch_bounds__(D) void agg_edges_kernel(
    const float* __restrict__ h, const long long* __restrict__ src,
    const long long* __restrict__ dst, const float* __restrict__ coef,
    float* __restrict__ out, int nE) {
  const int EPB = 16;  // edges per block
  const int d = threadIdx.x;
  const int e0 = blockIdx.x * EPB;
#pragma unroll 1
  for (int i = 0; i < EPB; ++i) {
    int e = e0 + i;
    if (e >= nE) return;  // uniform across the block
    int s = (int)src[e];
    int t = (int)dst[e];
    float c = coef[e];
    if (e + 1 < nE) {  // global_prefetch of next gather row
      __builtin_prefetch(&h[(size_t)((int)src[e + 1]) * D + d], 0, 0);
    }
    atomicAdd(&out[(size_t)t * D + d], h[(size_t)s * D + d] * c);
  }
}

// ---------------------------------------------------------------------------
// WMMA fp32 GEMM:  C[nrows,M] = act(A[nrows,K] @ Wpacked + biaspad)
// M is the padded column count (multiple of 16); ld of C == M.
// 256-thread block = 8 waves; each wave owns one 16-row tile and CT
// register-resident 16-col accumulator tiles. K stepped by 4 with
// v_wmma_f32_16x16x4_f32. Fully branch-free hot loop: B comes from the
// packed buffer (b64 loads at immediate offsets), A as contiguous v2f,
// stores unconditional into the padded C. EXEC is all-ones throughout
// (nrows % 16 == 0, wave-uniform early-out only).
// ---------------------------------------------------------------------------
template <int K, int M, int CT, bool RELU>
__global__ __launch_bounds__(256) void gemm_wmma_kernel(
    const float* __restrict__ A, const float* __restrict__ Bt,
    const float* __restrict__ bias, float* __restrict__ C, int nrows) {
  const int wave = threadIdx.x >> 5;
  const int lane = threadIdx.x & 31;
  const int row0 = (blockIdx.x * 8 + wave) * 16;
  if (row0 >= nrows) return;  // wave-uniform
  const int m = lane & 15;
  const int kh = lane >> 4;  // 0 or 1
  constexpr int K4 = K / 4;
  const float* Arow = A + (size_t)(row0 + m) * K + 2 * kh;

  for (int ct0 = 0; ct0 < M / 16; ct0 += CT) {
    v8f acc[CT];
    const float* bbase[CT];
#pragma unroll
    for (int c = 0; c < CT; ++c) {
      v8f z = {0.f, 0.f, 0.f, 0.f, 0.f, 0.f, 0.f, 0.f};
      acc[c] = z;
      bbase[c] = Bt + (size_t)(ct0 + c) * K4 * 64 + kh * 32 + m * 2;
    }
#pragma unroll
    for (int k4 = 0; k4 < K4; ++k4) {
      v2f a = *(const v2f*)(Arow + k4 * 4);
#pragma unroll
      for (int c = 0; c < CT; ++c) {
        v2f b = *(const v2f*)(bbase[c] + k4 * 64);
        acc[c] = __builtin_amdgcn_wmma_f32_16x16x4_f32(
            /*neg_a=*/false, a, /*neg_b=*/false, b,
            /*c_mod=*/(short)0, acc[c], /*reuse_a=*/false, /*reuse_b=*/false);
      }
    }
#pragma unroll
    for (int c = 0; c < CT; ++c) {
      int col = (ct0 + c) * 16 + m;
      float bv = bias[col];
#pragma unroll
      for (int i = 0; i < 8; ++i) {
        float v = acc[c][i] + bv;
        if (RELU) v = fmaxf(v, 0.0f);
        C[(size_t)(row0 + kh * 8 + i) * M + col] = v;
      }
    }
  }
}

// ---------------------------------------------------------------------------
// row-wise log_softmax over DOUT=40 logits (ld = DOUTP); one wave32 per row.
// out[(n*NLAYERS + layer)*DOUT + c] = logits[n,c] - logsumexp(row)
// ---------------------------------------------------------------------------
__global__ __launch_bounds__(256) void logsoftmax_kernel(
    const float* __restrict__ logits, float* __restrict__ out, int nrows,
    int layer) {
  const int wave = threadIdx.x >> 5;
  const int lane = threadIdx.x & 31;
  const int row = blockIdx.x * 8 + wave;
  if (row >= nrows) return;
  const float* p = logits + (size_t)row * DOUTP;
  float v0 = p[lane];  // lane 0..31 < 40 always valid
  bool has1 = (lane + 32) < DOUT;
  float v1 = has1 ? p[lane + 32] : -INFINITY;
  float mx = fmaxf(v0, v1);
#pragma unroll
  for (int off = 16; off > 0; off >>= 1)
    mx = fmaxf(mx, __shfl_xor(mx, off, 32));
  float s = __expf(v0 - mx) + (has1 ? __expf(v1 - mx) : 0.0f);
#pragma unroll
  for (int off = 16; off > 0; off >>= 1) s += __shfl_xor(s, off, 32);
  float lse = mx + __logf(s);
  float* o = out + ((size_t)row * NLAYERS + layer) * DOUT;
  o[lane] = v0 - lse;
  if (has1) o[lane + 32] = v1 - lse;
}

// ---------------------------------------------------------------------------
static inline size_t alignup(size_t x) { return (x + 255) & ~(size_t)255; }

extern "C" void kernel_launch(void* const* d_in, const int* in_sizes, int n_in,
                              void* d_out, int out_size, void* d_ws,
                              size_t ws_size, hipStream_t stream) {
  const float* x = (const float*)d_in[0];
  const long long* ei = (const long long*)d_in[1];  // int64 edge_index [2,E]
  const float* Wc0 = (const float*)d_in[2];
  const float* bc0 = (const float*)d_in[3];
  const float* Wc1 = (const float*)d_in[4];
  const float* bc1 = (const float*)d_in[5];
  // Wc2/bc2 (d_in[6],d_in[7]) feed a value the reference discards -> skipped
  const float* We0 = (const float*)d_in[8];
  const float* be0 = (const float*)d_in[9];
  const float* We1 = (const float*)d_in[10];
  const float* be1 = (const float*)d_in[11];
  const float* We2 = (const float*)d_in[12];
  const float* be2 = (const float*)d_in[13];
  const float* We3 = (const float*)d_in[14];
  const float* be3 = (const float*)d_in[15];
  float* out = (float*)d_out;

  const int nN = in_sizes[0] / DIN;  // 100000
  const int nE = in_sizes[1] / 2;    // 1600000
  const long long* src = ei;
  const long long* dst = ei + nE;

  // workspace carve-up
  char* ws = (char*)d_ws;
  size_t off = 0;
  float* dinv = (float*)(ws + off);    off += alignup((size_t)nN * 4);
  float* coef = (float*)(ws + off);    off += alignup((size_t)nE * 4);
  float* bufA = (float*)(ws + off);    off += alignup((size_t)nN * DH * 4);
  float* bufB = (float*)(ws + off);    off += alignup((size_t)nN * DH * 4);
  float* logits = (float*)(ws + off);  off += alignup((size_t)nN * DOUTP * 4);
  float* tWe0 = (float*)(ws + off);    off += alignup((size_t)DIN * DOUTP * 4);
  float* tWe1 = (float*)(ws + off);    off += alignup((size_t)DIN * DOUTP * 4);
  float* tWe2 = (float*)(ws + off);    off += alignup((size_t)DH * DOUTP * 4);
  float* tWe3 = (float*)(ws + off);    off += alignup((size_t)DH * DOUTP * 4);
  float* tWc0 = (float*)(ws + off);    off += alignup((size_t)DIN * DH * 4);
  float* tWc1 = (float*)(ws + off);    off += alignup((size_t)DH * DH * 4);
  float* pbe0 = (float*)(ws + off);    off += alignup(DOUTP * 4);
  float* pbe1 = (float*)(ws + off);    off += alignup(DOUTP * 4);
  float* pbe2 = (float*)(ws + off);    off += alignup(DOUTP * 4);
  float* pbe3 = (float*)(ws + off);    off += alignup(DOUTP * 4);
  (void)ws_size;

  const int T = 256;
  const int gemmBlocks = (nN + 127) / 128;  // 8 waves * 16 rows / block
  const int lsmBlocks = (nN + 7) / 8;       // 8 rows / block
  const int edgeBlocks = (nE + 15) / 16;    // 16 edges / block

  // --- weight / bias packing (tiny, once per launch) ---------------------
  packB_kernel<<<(DIN * DOUTP + T - 1) / T, T, 0, stream>>>(We0, tWe0, DIN, DOUT, DOUTP);
  packB_kernel<<<(DIN * DOUTP + T - 1) / T, T, 0, stream>>>(We1, tWe1, DIN, DOUT, DOUTP);
  packB_kernel<<<(DH * DOUTP + T - 1) / T, T, 0, stream>>>(We2, tWe2, DH, DOUT, DOUTP);
  packB_kernel<<<(DH * DOUTP + T - 1) / T, T, 0, stream>>>(We3, tWe3, DH, DOUT, DOUTP);
  packB_kernel<<<(DIN * DH + T - 1) / T, T, 0, stream>>>(Wc0, tWc0, DIN, DH, DH);
  packB_kernel<<<(DH * DH + T - 1) / T, T, 0, stream>>>(Wc1, tWc1, DH, DH, DH);
  padbias_kernel<<<1, DOUTP, 0, stream>>>(be0, pbe0, DOUT, DOUTP);
  padbias_kernel<<<1, DOUTP, 0, stream>>>(be1, pbe1, DOUT, DOUTP);
  padbias_kernel<<<1, DOUTP, 0, stream>>>(be2, pbe2, DOUT, DOUTP);
  padbias_kernel<<<1, DOUTP, 0, stream>>>(be3, pbe3, DOUT, DOUTP);

  // --- degree / normalization coefficients -------------------------------
  zero_kernel<<<(nN + T - 1) / T, T, 0, stream>>>(dinv, nN);
  deg_kernel<<<(nE + T - 1) / T, T, 0, stream>>>(dst, dinv, nE);
  dinv_kernel<<<(nN + T - 1) / T, T, 0, stream>>>(dinv, nN);
  coef_kernel<<<(nE + T - 1) / T, T, 0, stream>>>(src, dst, dinv, coef, nE);

  // --- layer 0 head: log_softmax(x @ We0 + be0) --------------------------
  gemm_wmma_kernel<DIN, DOUTP, 3, false>
      <<<gemmBlocks, T, 0, stream>>>(x, tWe0, pbe0, logits, nN);
  logsoftmax_kernel<<<lsmBlocks, T, 0, stream>>>(logits, out, nN, 0);

  // --- i = 0 : h = agg(x) [128-dim] --------------------------------------
  agg_init_kernel<<<((size_t)nN * DIN + T - 1) / T, T, 0, stream>>>(
      x, dinv, bufA, nN * DIN, 7);
  agg_edges_kernel<DIN>
      <<<edgeBlocks, DIN, 0, stream>>>(x, src, dst, coef, bufA, nE);
  gemm_wmma_kernel<DIN, DOUTP, 3, false>
      <<<gemmBlocks, T, 0, stream>>>(bufA, tWe1, pbe1, logits, nN);
  logsoftmax_kernel<<<lsmBlocks, T, 0, stream>>>(logits, out, nN, 1);
  gemm_wmma_kernel<DIN, DH, 4, true>
      <<<gemmBlocks, T, 0, stream>>>(bufA, tWc0, bc0, bufB, nN);

  // --- i = 1 : h = agg(h) [256-dim] --------------------------------------
  agg_init_kernel<<<((size_t)nN * DH + T - 1) / T, T, 0, stream>>>(
      bufB, dinv, bufA, nN * DH, 8);
  agg_edges_kernel<DH>
      <<<edgeBlocks, DH, 0, stream>>>(bufB, src, dst, coef, bufA, nE);
  gemm_wmma_kernel<DH, DOUTP, 3, false>
      <<<gemmBlocks, T, 0, stream>>>(bufA, tWe2, pbe2, logits, nN);
  logsoftmax_kernel<<<lsmBlocks, T, 0, stream>>>(logits, out, nN, 2);
  gemm_wmma_kernel<DH, DH, 4, true>
      <<<gemmBlocks, T, 0, stream>>>(bufA, tWc1, bc1, bufB, nN);

  // --- i = 2 : h = agg(h), final head only (h@Wc2 is dead in reference) --
  agg_init_kernel<<<((size_t)nN * DH + T - 1) / T, T, 0, stream>>>(
      bufB, dinv, bufA, nN * DH, 8);
  agg_edges_kernel<DH>
      <<<edgeBlocks, DH, 0, stream>>>(bufB, src, dst, coef, bufA, nE);
  gemm_wmma_kernel<DH, DOUTP, 3, false>
      <<<gemmBlocks, T, 0, stream>>>(bufA, tWe3, pbe3, logits, nN);
  logsoftmax_kernel<<<lsmBlocks, T, 0, stream>>>(logits, out, nN, 3);
}